// HeteroRGCN_7129645711537
// MI455X (gfx1250) — compile-verified
//
#include <hip/hip_runtime.h>

// ---------------------------------------------------------------------------
// HeteroRGCN (2-layer) for MI455X / gfx1250, wave32, WMMA fp32 path.
//
// GEMMs use V_WMMA_F32_16X16X4_F32 (full fp32 fidelity, K=128 in fp32).
// Weights are pre-packed into the WMMA B-register layout so every B fetch is
// one coalesced global_load_b64; each wave computes 32 rows (two 16-row M
// tiles) sharing the B registers -> ~16 WMMA per 10 b64 loads in the k-loop.
// Edge mean-aggregation uses L2-resident global_atomic_add_f32: message
// buffers (51.2MB / 25.6MB) fit in the 192MB L2, so scatter atomics and
// random gathers resolve in L2. Streaming traffic ~1GB -> ~50us floor at
// 23.3 TB/s; fp32 WMMA FLOPs (~20GF) are negligible against that.
// ---------------------------------------------------------------------------

typedef float v2f __attribute__((ext_vector_type(2)));
typedef float v8f __attribute__((ext_vector_type(8)));

#define NUSER 100000
#define NITEM 100000
#define NEDGE 500000

// Pack W[128 x N] into WMMA B-layout: for k-quad kq, tile nt, lane L:
//   P[((kq*NT)+nt)*32 + L] = { W[kq*4 + 2*(L>=16) + 0][nt*16 + L%16],
//                              W[kq*4 + 2*(L>=16) + 1][nt*16 + L%16] }  (float2)
template <int N>
__global__ void repack_W(const float* __restrict__ W, float* __restrict__ P) {
  constexpr int NT = N / 16;
  const int idx = blockIdx.x * blockDim.x + threadIdx.x;
  if (idx >= 32 * NT * 32) return;  // 32 k-quads * NT tiles * 32 lanes
  const int lane = idx & 31;
  const int nt = (idx >> 5) % NT;
  const int kq = (idx >> 5) / NT;
  const int nIdx = lane & 15;
  const int hi = lane >> 4;
  const int ka = kq * 4 + 2 * hi;
  const int n = nt * 16 + nIdx;
  float2 b;
  b.x = W[(size_t)(ka + 0) * N + n];
  b.y = W[(size_t)(ka + 1) * N + n];
  *(float2*)(P + (size_t)idx * 2) = b;
}

// C[M x N] = A[M x 128] * W[128 x N] + bias   (K = 128, N = 128 or 64)
// One wave -> 32 rows (two 16-row WMMA tiles) x all N columns; B registers
// shared across both row tiles. Wp is the packed B layout from repack_W.
template <int N>
__global__ __launch_bounds__(256) void gemm_k128_wmma(
    const float* __restrict__ A, const float* __restrict__ Wp,
    const float* __restrict__ bias, float* __restrict__ C, int M) {
  constexpr int NT = N / 16;
  const int lane = threadIdx.x & 31;
  const int waveId = (blockIdx.x * blockDim.x + threadIdx.x) >> 5;
  const int m0 = waveId * 32;
  if (m0 >= M) return;  // wave-uniform exit: EXEC stays all-ones for WMMA
  const int nIdx = lane & 15;  // N column within tile / M row for A loads
  const int hi = lane >> 4;    // K-pair selector (A/B), row-half (C/D)

  v8f acc0[NT], acc1[NT];
#pragma unroll
  for (int nt = 0; nt < NT; ++nt) {
    v8f z = {0.f, 0.f, 0.f, 0.f, 0.f, 0.f, 0.f, 0.f};
    acc0[nt] = z;
    acc1[nt] = z;
  }

  const float* arow0 = A + (size_t)(m0 + nIdx) * 128;
  const float* arow1 = arow0 + (size_t)16 * 128;
#pragma unroll 4
  for (int k0 = 0; k0 < 128; k0 += 4) {
    const int ka = k0 + 2 * hi;
    v2f a0 = *(const v2f*)(arow0 + ka);
    v2f a1 = *(const v2f*)(arow1 + ka);
    const float* bbase = Wp + (((size_t)(k0 >> 2) * NT) << 6) + (lane << 1);
#pragma unroll
    for (int nt = 0; nt < NT; ++nt) {
      v2f b = *(const v2f*)(bbase + ((size_t)nt << 6));  // coalesced b64
      acc0[nt] = __builtin_amdgcn_wmma_f32_16x16x4_f32(
          false, a0, false, b, (short)0, acc0[nt], false, false);
      acc1[nt] = __builtin_amdgcn_wmma_f32_16x16x4_f32(
          false, a1, false, b, (short)0, acc1[nt], false, false);
    }
  }

#pragma unroll
  for (int nt = 0; nt < NT; ++nt) {
    const float bv = bias[nt * 16 + nIdx];
#pragma unroll
    for (int r = 0; r < 8; ++r) {
      const int m = m0 + r + 8 * hi;
      C[(size_t)m * N + nt * 16 + nIdx] = acc0[nt][r] + bv;
      C[(size_t)(m + 16) * N + nt * 16 + nIdx] = acc1[nt][r] + bv;
    }
  }
}

// Zero fill (grid-stride)
__global__ void zero_kernel(float* __restrict__ p, int n) {
  int i = blockIdx.x * blockDim.x + threadIdx.x;
  const int stride = gridDim.x * blockDim.x;
  for (; i < n; i += stride) p[i] = 0.0f;
}

// Segment-sum of gathered messages: one wave per edge, lane moves F/32 floats.
// acc (51.2/25.6 MB) is L2-resident -> atomics resolve in L2 atomic units.
template <int F>
__global__ __launch_bounds__(256) void edge_agg(
    const float* __restrict__ msg, const int* __restrict__ src,
    const int* __restrict__ dst, float* __restrict__ acc,
    float* __restrict__ cnt, int E) {
  const int lane = threadIdx.x & 31;
  const int wave = (blockIdx.x * blockDim.x + threadIdx.x) >> 5;
  const int nwaves = (gridDim.x * blockDim.x) >> 5;
  for (int e = wave; e < E; e += nwaves) {
    const int s = src[e];
    const int d = dst[e];
    if (F == 128) {
      const float4 v = *(const float4*)(msg + (size_t)s * 128 + lane * 4);
      float* p = acc + (size_t)d * 128 + lane * 4;
      atomicAdd(p + 0, v.x);
      atomicAdd(p + 1, v.y);
      atomicAdd(p + 2, v.z);
      atomicAdd(p + 3, v.w);
    } else {
      const float2 v = *(const float2*)(msg + (size_t)s * 64 + lane * 2);
      float* p = acc + (size_t)d * 64 + lane * 2;
      atomicAdd(p + 0, v.x);
      atomicAdd(p + 1, v.y);
    }
    if (lane == 0) atomicAdd(cnt + d, 1.0f);
  }
}

// out = maybe_lrelu( (cnt>0 ? acc/cnt : 0) + h0 )
__global__ void finalize_kernel(const float* __restrict__ acc,
                                const float* __restrict__ cnt,
                                const float* __restrict__ h0,
                                float* __restrict__ out, int nElem,
                                int fShift /*log2(F)*/, int doLrelu) {
  const int i = blockIdx.x * blockDim.x + threadIdx.x;
  if (i >= nElem) return;
  const float c = cnt[i >> fShift];
  float v = (c > 0.0f) ? acc[i] / c : 0.0f;
  v += h0[i];
  if (doLrelu) v = (v > 0.0f) ? v : 0.01f * v;
  out[i] = v;
}

extern "C" void kernel_launch(void* const* d_in, const int* in_sizes, int n_in,
                              void* d_out, int out_size, void* d_ws,
                              size_t ws_size, hipStream_t stream) {
  const float* emb_user = (const float*)d_in[0];
  const float* emb_item = (const float*)d_in[1];
  const float* W1_0 = (const float*)d_in[2];
  const float* b1_0 = (const float*)d_in[3];
  const float* W1_ui = (const float*)d_in[4];
  const float* b1_ui = (const float*)d_in[5];
  const float* W1_iu = (const float*)d_in[6];
  const float* b1_iu = (const float*)d_in[7];
  const float* W2_0 = (const float*)d_in[8];
  const float* b2_0 = (const float*)d_in[9];
  const float* W2_ui = (const float*)d_in[10];
  const float* b2_ui = (const float*)d_in[11];
  const float* W2_iu = (const float*)d_in[12];
  const float* b2_iu = (const float*)d_in[13];
  const int* src_ui = (const int*)d_in[14];
  const int* dst_ui = (const int*)d_in[15];
  const int* src_iu = (const int*)d_in[16];
  const int* dst_iu = (const int*)d_in[17];
  float* out = (float*)d_out;  // [h_u (100000x64) | h_i (100000x64)]

  const size_t NODE128 = (size_t)NUSER * 128;  // 12.8M floats
  const size_t NODE64 = (size_t)NUSER * 64;    // 6.4M floats
  char* ws = (char*)d_ws;
  float* bufA = (float*)(ws);                 // h0_u -> h1_u
  float* bufB = (float*)(ws + NODE128 * 4);   // wh_u ; L2: h0o_u + who_u
  float* bufC = (float*)(ws + NODE128 * 8);   // h0_i -> h1_i
  float* bufD = (float*)(ws + NODE128 * 12);  // wh_i ; L2: h0o_i + who_i
  float* accB = (float*)(ws + NODE128 * 16);  // shared accumulator
  float* cntB = (float*)(ws + NODE128 * 20);  // 100000 counts
  // packed weights (after counts, 1MB pad): 3 x 64KB + 3 x 32KB
  char* wsp = ws + NODE128 * 20 + (1 << 20);
  float* P1_0 = (float*)(wsp);
  float* P1_ui = (float*)(wsp + 64 * 1024);
  float* P1_iu = (float*)(wsp + 128 * 1024);
  float* P2_0 = (float*)(wsp + 192 * 1024);
  float* P2_ui = (float*)(wsp + 224 * 1024);
  float* P2_iu = (float*)(wsp + 256 * 1024);

  float* h0o_u = bufB;
  float* who_u = bufB + NODE64;
  float* h0o_i = bufD;
  float* who_i = bufD + NODE64;

  const dim3 blk(256);
  const int mtiles32 = NUSER / 32;              // 3125 waves (32 rows each)
  const int gemmBlocks = (mtiles32 + 7) / 8;    // 8 waves / block
  const int aggBlocks = 1024;
  const int fin128Blocks = (int)((NODE128 + 255) / 256);
  const int fin64Blocks = (int)((NODE64 + 255) / 256);

  // Pre-pack all six weight matrices into WMMA B-layout (tiny kernels).
  repack_W<128><<<32, blk, 0, stream>>>(W1_0, P1_0);
  repack_W<128><<<32, blk, 0, stream>>>(W1_ui, P1_ui);
  repack_W<128><<<32, blk, 0, stream>>>(W1_iu, P1_iu);
  repack_W<64><<<16, blk, 0, stream>>>(W2_0, P2_0);
  repack_W<64><<<16, blk, 0, stream>>>(W2_ui, P2_ui);
  repack_W<64><<<16, blk, 0, stream>>>(W2_iu, P2_iu);

  // ---------------- Layer 1 (EMB=128 -> HID=128) ----------------
  gemm_k128_wmma<128><<<gemmBlocks, blk, 0, stream>>>(emb_user, P1_0, b1_0, bufA, NUSER);
  gemm_k128_wmma<128><<<gemmBlocks, blk, 0, stream>>>(emb_user, P1_ui, b1_ui, bufB, NUSER);
  gemm_k128_wmma<128><<<gemmBlocks, blk, 0, stream>>>(emb_item, P1_0, b1_0, bufC, NITEM);
  gemm_k128_wmma<128><<<gemmBlocks, blk, 0, stream>>>(emb_item, P1_iu, b1_iu, bufD, NITEM);

  // user->item mean agg, then h1_i = lrelu(mean + h0_i) (in place over bufC)
  zero_kernel<<<2048, blk, 0, stream>>>(accB, (int)NODE128);
  zero_kernel<<<256, blk, 0, stream>>>(cntB, NITEM);
  edge_agg<128><<<aggBlocks, blk, 0, stream>>>(bufB, src_ui, dst_ui, accB, cntB, NEDGE);
  finalize_kernel<<<fin128Blocks, blk, 0, stream>>>(accB, cntB, bufC, bufC, (int)NODE128, 7, 1);

  // item->user mean agg, then h1_u = lrelu(mean + h0_u) (in place over bufA)
  zero_kernel<<<2048, blk, 0, stream>>>(accB, (int)NODE128);
  zero_kernel<<<256, blk, 0, stream>>>(cntB, NUSER);
  edge_agg<128><<<aggBlocks, blk, 0, stream>>>(bufD, src_iu, dst_iu, accB, cntB, NEDGE);
  finalize_kernel<<<fin128Blocks, blk, 0, stream>>>(accB, cntB, bufA, bufA, (int)NODE128, 7, 1);

  // ---------------- Layer 2 (HID=128 -> OUT=64) ----------------
  gemm_k128_wmma<64><<<gemmBlocks, blk, 0, stream>>>(bufA, P2_0, b2_0, h0o_u, NUSER);
  gemm_k128_wmma<64><<<gemmBlocks, blk, 0, stream>>>(bufA, P2_ui, b2_ui, who_u, NUSER);
  gemm_k128_wmma<64><<<gemmBlocks, blk, 0, stream>>>(bufC, P2_0, b2_0, h0o_i, NITEM);
  gemm_k128_wmma<64><<<gemmBlocks, blk, 0, stream>>>(bufC, P2_iu, b2_iu, who_i, NITEM);

  // user->item: h_i = mean + h0o_i  -> out[NODE64 ..]
  zero_kernel<<<2048, blk, 0, stream>>>(accB, (int)NODE64);
  zero_kernel<<<256, blk, 0, stream>>>(cntB, NITEM);
  edge_agg<64><<<aggBlocks, blk, 0, stream>>>(who_u, src_ui, dst_ui, accB, cntB, NEDGE);
  finalize_kernel<<<fin64Blocks, blk, 0, stream>>>(accB, cntB, h0o_i, out + NODE64, (int)NODE64, 6, 0);

  // item->user: h_u = mean + h0o_u  -> out[0 ..]
  zero_kernel<<<2048, blk, 0, stream>>>(accB, (int)NODE64);
  zero_kernel<<<256, blk, 0, stream>>>(cntB, NUSER);
  edge_agg<64><<<aggBlocks, blk, 0, stream>>>(who_i, src_iu, dst_iu, accB, cntB, NEDGE);
  finalize_kernel<<<fin64Blocks, blk, 0, stream>>>(accB, cntB, h0o_u, out, (int)NODE64, 6, 0);

  (void)in_sizes; (void)n_in; (void)out_size; (void)ws_size;
}